// PDMEncodeur_seq_14216341750164
// MI455X (gfx1250) — compile-verified
//
#include <hip/hip_runtime.h>

typedef float v2f   __attribute__((ext_vector_type(2)));
typedef float v8f   __attribute__((ext_vector_type(8)));
typedef float f32x4 __attribute__((ext_vector_type(4)));

// Four accumulated V_WMMA_F32_16X16X4_F32: S = L * X, L = 16x16 lower-tri ones,
// X = 16 timesteps x 16 batch rows (x = wav/2 + 0.5 applied at load).
// A layout (16x4 f32): lane holds M=col; VGPR0 -> K=4j+2*half, VGPR1 -> K=4j+1+2*half.
// B layout (4x16 f32): VGPR0 -> K=4j+half, VGPR1 -> K=4j+2+half, N=col.
__device__ __forceinline__ v8f prefix_wmma(const float* __restrict__ p,
                                           const v2f* A, int half) {
  v8f acc = {};
#pragma unroll
  for (int j = 0; j < 4; ++j) {
    const float b0 = p[4 * j + half]     * 0.5f + 0.5f;
    const float b1 = p[4 * j + 2 + half] * 0.5f + 0.5f;
    v2f Bv; Bv.x = b0; Bv.y = b1;
    acc = __builtin_amdgcn_wmma_f32_16x16x4_f32(
        /*neg_a=*/false, A[j], /*neg_b=*/false, Bv,
        /*c_mod=*/(short)0, acc, /*reuse_a=*/false, /*reuse_b=*/false);
  }
  return acc;
}

// One wave per block; block owns 32 batch rows (two 16-row WMMA tiles).
// Lower lanes own rows g1 (tile 1), upper lanes own rows g2 (tile 2).
// One SWAPX16 swizzle per D-VGPR (off the carry chain) gives each lane all 16
// prefix sums of its own row; the delta-sigma chase then runs entirely
// lane-local with a 2-op/step dependence chain.
__global__ __launch_bounds__(32) void pdm_encode_kernel(
    const float* __restrict__ wav, float* __restrict__ out, int T) {
  const int lane = threadIdx.x & 31;
  const int col  = lane & 15;
  const int half = lane >> 4;

  const long g1base  = (long)(blockIdx.x * 32 + col) * (long)T;   // tile-1 row feed
  const long g2base  = g1base + 16L * (long)T;                    // tile-2 row feed
  const long ownbase = (long)(blockIdx.x * 32 + 16 * half + col) * (long)T;

  v2f A[4];
#pragma unroll
  for (int j = 0; j < 4; ++j) {
    const int k0 = 4 * j + 2 * half;
    A[j].x = (k0     <= col) ? 1.0f : 0.0f;
    A[j].y = (k0 + 1 <= col) ? 1.0f : 0.0f;
  }

  const int nchunk = T >> 4;
  float Slo[8], Shi[8];    // this chunk's prefix sums for this lane's row
  float nSlo[8], nShi[8];  // next chunk's (software pipeline)

  // ---- produce chunk 0 ----
  {
    const v8f a1 = prefix_wmma(wav + g1base, A, half);
    const v8f a2 = prefix_wmma(wav + g2base, A, half);
#pragma unroll
    for (int v = 0; v < 8; ++v) {
      const float send = half ? a1[v] : a2[v];
      const float got  = __int_as_float(
          __builtin_amdgcn_ds_swizzle(__float_as_int(send), 0x401F));
      Slo[v] = half ? got   : a1[v];
      Shi[v] = half ? a2[v] : got;
    }
  }

  float e = 0.0f;  // per-lane delta-sigma error carry (own row)

  for (int c = 0; c < nchunk; ++c) {
    const int t0  = c << 4;
    const int t0n = (c + 1 < nchunk) ? t0 + 16 : 0;  // clamp: last iter harmless

    // Stream ahead for both row groups.
    __builtin_prefetch(&wav[g1base + t0 + 512], 0, 1);
    __builtin_prefetch(&wav[g2base + t0 + 512], 0, 1);

    // ---- produce chunk c+1 (independent: hides under the chase chain) ----
    {
      const v8f a1 = prefix_wmma(wav + g1base + t0n, A, half);
      const v8f a2 = prefix_wmma(wav + g2base + t0n, A, half);
#pragma unroll
      for (int v = 0; v < 8; ++v) {
        const float send = half ? a1[v] : a2[v];
        const float got  = __int_as_float(
            __builtin_amdgcn_ds_swizzle(__float_as_int(send), 0x401F));
        nSlo[v] = half ? got   : a1[v];
        nShi[v] = half ? a2[v] : got;
      }
    }

    // ---- chase chunk c: thresholds are independent, chain is cmp+cndmask ----
    float thr[16];
#pragma unroll
    for (int v = 0; v < 8; ++v) { thr[v] = e + Slo[v]; thr[8 + v] = e + Shi[v]; }

    float cs[17];
    cs[0] = 0.0f;
#pragma unroll
    for (int v = 0; v < 16; ++v) {
      const float c1 = cs[v] + 1.0f;              // off-chain (parallel w/ cmp)
      cs[v + 1] = (thr[v] > cs[v]) ? c1 : cs[v];  // 2-op serial step
    }
    e = thr[15] - cs[16];                          // err carry to next chunk

    float s[16];
#pragma unroll
    for (int v = 0; v < 16; ++v) s[v] = cs[v + 1] - cs[v];  // parallel recover

    // ---- store 16 spikes: four b128 stores, 16B-aligned (t0 % 16 == 0) ----
    float* op = out + ownbase + t0;
#pragma unroll
    for (int q = 0; q < 4; ++q) {
      f32x4 o;
      o.x = s[4 * q + 0]; o.y = s[4 * q + 1];
      o.z = s[4 * q + 2]; o.w = s[4 * q + 3];
      *(f32x4*)(op + 4 * q) = o;
    }

    // rotate pipeline
#pragma unroll
    for (int v = 0; v < 8; ++v) { Slo[v] = nSlo[v]; Shi[v] = nShi[v]; }
  }
}

extern "C" void kernel_launch(void* const* d_in, const int* in_sizes, int n_in,
                              void* d_out, int out_size, void* d_ws, size_t ws_size,
                              hipStream_t stream) {
  const float* wav = (const float*)d_in[0];
  float* out = (float*)d_out;
  const int T = 160000;            // [B, T] per reference
  const int B = in_sizes[0] / T;   // 128
  dim3 grid(B / 32), block(32);    // one wave32 per block, 32 rows each
  hipLaunchKernelGGL(pdm_encode_kernel, grid, block, 0, stream, wav, out, T);
}